// GNN_84043920048593
// MI455X (gfx1250) — compile-verified
//
#include <hip/hip_runtime.h>
#include <hip/hip_bf16.h>

typedef __attribute__((ext_vector_type(2))) float v2f;
typedef __attribute__((ext_vector_type(8))) float v8f;

// ---------------------------------------------------------------------------
// Edge scatter: mean-aggregation numerator + degree counts.
// 16 lanes per edge, each lane moves a float4 slice of the 64-wide feature.
// Accumulators live in L2 (192MB on MI455X easily holds ~104MB of scratch),
// so global_atomic_add_f32 resolves on-chip.
// ---------------------------------------------------------------------------
__global__ void gnn_edge_scatter(const float* __restrict__ xsrc,
                                 const int* __restrict__ src,
                                 const int* __restrict__ dst,
                                 const int* __restrict__ remap, // nullptr or label_node_id
                                 float* __restrict__ agg,
                                 float* __restrict__ cnt,
                                 long long E) {
  long long total  = E << 4;
  long long stride = (long long)gridDim.x * blockDim.x;
  for (long long tid = (long long)blockIdx.x * blockDim.x + threadIdx.x;
       tid < total; tid += stride) {
    long long e = tid >> 4;
    int g = (int)(tid & 15);
    int s = src[e];
    int d = dst[e];
    if (remap) s = remap[s];
    const float4 v = *(const float4*)(xsrc + (long long)s * 64 + g * 4);
    float* p = agg + (long long)d * 64 + g * 4;
    atomicAdd(p + 0, v.x);
    atomicAdd(p + 1, v.y);
    atomicAdd(p + 2, v.z);
    atomicAdd(p + 3, v.w);
    if (g == 0) atomicAdd(cnt + d, 1.0f);
  }
}

// ---------------------------------------------------------------------------
// WMMA fragment helpers (V_WMMA_F32_16X16X4_F32 layouts, ISA 7.12.2):
//   A 16x4 f32 : lanes 0-15 -> row M=lane,  K = {k0, k0+1}
//                lanes 16-31 -> row M=lane-16, K = {k0+2, k0+3}
//   B 4x16 f32 : lanes 0-15 -> col N=lane,  K = {k0, k0+1}   (B[k][n] = W[n][k])
//   C/D 16x16  : vgpr v -> M = v + 8*hi, N = lane&15
// ---------------------------------------------------------------------------
__device__ __forceinline__ v2f load_a_frag(const float* __restrict__ rowp, int k0,
                                           float scale) {
  v2f a;
  a.x = rowp[k0] * scale;
  a.y = rowp[k0 + 1] * scale;
  return a;
}

__device__ __forceinline__ v2f load_b_frag(const float* __restrict__ W, int n, int k0) {
  v2f b;
  b.x = W[n * 64 + k0];
  b.y = W[n * 64 + k0 + 1];
  return b;
}

// ---------------------------------------------------------------------------
// Label output: relu( agg_tl/cnt @ Wl_tl^T + x @ (Wr_tl+Wr_ll)^T
//                   + agg_ll/cnt @ Wl_ll^T + bl_tl + bl_ll )
// One wave per 16-row tile; 4 column tiles of 16; K=64 in steps of 4.
// ---------------------------------------------------------------------------
__global__ void gnn_label_out(const float* __restrict__ label_embed,
                              const int* __restrict__ label_node_id,
                              const float* __restrict__ agg_tl,
                              const float* __restrict__ cnt_tl,
                              const float* __restrict__ agg_ll,
                              const float* __restrict__ cnt_ll,
                              const float* __restrict__ Wl_tl,
                              const float* __restrict__ bl_tl,
                              const float* __restrict__ Wr_tl,
                              const float* __restrict__ Wl_ll,
                              const float* __restrict__ bl_ll,
                              const float* __restrict__ Wr_ll,
                              float* __restrict__ out, int NL) {
  const int wave = threadIdx.x >> 5;
  const int lane = threadIdx.x & 31;
  const int tile = blockIdx.x * (blockDim.x >> 5) + wave;
  if (tile * 16 >= NL) return;           // wave-uniform: EXEC stays all-ones

  const int m0 = tile * 16;
  const int lr = lane & 15;
  const int hi = lane >> 4;
  const int row = m0 + lr;

  const float inv_tl = 1.0f / fmaxf(cnt_tl[row], 1.0f);
  const float inv_ll = 1.0f / fmaxf(cnt_ll[row], 1.0f);
  const int   xrow   = label_node_id[row];
  const float* xa  = label_embed + (long long)xrow * 64;
  const float* atl = agg_tl + (long long)row * 64;
  const float* all_ = agg_ll + (long long)row * 64;

  v8f c[4] = {};

  for (int ks = 0; ks < 16; ++ks) {
    const int k0 = ks * 4 + hi * 2;
    const v2f A1 = load_a_frag(atl, k0, inv_tl);
    const v2f A2 = load_a_frag(xa, k0, 1.0f);
    const v2f A3 = load_a_frag(all_, k0, inv_ll);
#pragma unroll
    for (int nt = 0; nt < 4; ++nt) {
      const int n = nt * 16 + lr;
      const v2f B1 = load_b_frag(Wl_tl, n, k0);
      v2f B2;
      B2.x = Wr_tl[n * 64 + k0] + Wr_ll[n * 64 + k0];
      B2.y = Wr_tl[n * 64 + k0 + 1] + Wr_ll[n * 64 + k0 + 1];
      const v2f B3 = load_b_frag(Wl_ll, n, k0);
      c[nt] = __builtin_amdgcn_wmma_f32_16x16x4_f32(false, A1, false, B1,
                                                    (short)0, c[nt], false, false);
      c[nt] = __builtin_amdgcn_wmma_f32_16x16x4_f32(false, A2, false, B2,
                                                    (short)0, c[nt], false, false);
      c[nt] = __builtin_amdgcn_wmma_f32_16x16x4_f32(false, A3, false, B3,
                                                    (short)0, c[nt], false, false);
    }
  }

#pragma unroll
  for (int nt = 0; nt < 4; ++nt) {
    const int n = nt * 16 + lr;
    const float bias = bl_tl[n] + bl_ll[n];
#pragma unroll
    for (int v = 0; v < 8; ++v) {
      const int M = v + hi * 8;
      out[(long long)(m0 + M) * 64 + n] = fmaxf(c[nt][v] + bias, 0.0f);
    }
  }
}

// ---------------------------------------------------------------------------
// Title output: relu( agg_lt/cnt @ Wl_tl^T + title_x @ Wr_tl^T + bl_tl )
// ---------------------------------------------------------------------------
__global__ void gnn_title_out(const float* __restrict__ title_x,
                              const float* __restrict__ agg_lt,
                              const float* __restrict__ cnt_lt,
                              const float* __restrict__ Wl_tl,
                              const float* __restrict__ bl_tl,
                              const float* __restrict__ Wr_tl,
                              float* __restrict__ out, int NT) {
  const int wave = threadIdx.x >> 5;
  const int lane = threadIdx.x & 31;
  const int tile = blockIdx.x * (blockDim.x >> 5) + wave;
  if (tile * 16 >= NT) return;           // wave-uniform

  const int m0 = tile * 16;
  const int lr = lane & 15;
  const int hi = lane >> 4;
  const int row = m0 + lr;

  const float inv_lt = 1.0f / fmaxf(cnt_lt[row], 1.0f);
  const float* xa  = title_x + (long long)row * 64;
  const float* alt = agg_lt + (long long)row * 64;

  v8f c[4] = {};

  for (int ks = 0; ks < 16; ++ks) {
    const int k0 = ks * 4 + hi * 2;
    const v2f A1 = load_a_frag(alt, k0, inv_lt);
    const v2f A2 = load_a_frag(xa, k0, 1.0f);
#pragma unroll
    for (int nt = 0; nt < 4; ++nt) {
      const int n = nt * 16 + lr;
      const v2f B1 = load_b_frag(Wl_tl, n, k0);
      const v2f B2 = load_b_frag(Wr_tl, n, k0);
      c[nt] = __builtin_amdgcn_wmma_f32_16x16x4_f32(false, A1, false, B1,
                                                    (short)0, c[nt], false, false);
      c[nt] = __builtin_amdgcn_wmma_f32_16x16x4_f32(false, A2, false, B2,
                                                    (short)0, c[nt], false, false);
    }
  }

#pragma unroll
  for (int nt = 0; nt < 4; ++nt) {
    const int n = nt * 16 + lr;
    const float bias = bl_tl[n];
#pragma unroll
    for (int v = 0; v < 8; ++v) {
      const int M = v + hi * 8;
      out[(long long)(m0 + M) * 64 + n] = fmaxf(c[nt][v] + bias, 0.0f);
    }
  }
}

extern "C" void kernel_launch(void* const* d_in, const int* in_sizes, int n_in,
                              void* d_out, int out_size, void* d_ws, size_t ws_size,
                              hipStream_t stream) {
  const float* title_x       = (const float*)d_in[0];
  const float* label_embed   = (const float*)d_in[1];
  const float* Wl_tl         = (const float*)d_in[2];
  const float* bl_tl         = (const float*)d_in[3];
  const float* Wr_tl         = (const float*)d_in[4];
  const float* Wl_ll         = (const float*)d_in[5];
  const float* bl_ll         = (const float*)d_in[6];
  const float* Wr_ll         = (const float*)d_in[7];
  const int*   label_node_id = (const int*)d_in[8];
  const int*   tl_src        = (const int*)d_in[9];
  const int*   tl_dst        = (const int*)d_in[10];
  const int*   lt_src        = (const int*)d_in[11];
  const int*   lt_dst        = (const int*)d_in[12];
  const int*   ll_src        = (const int*)d_in[13];
  const int*   ll_dst        = (const int*)d_in[14];

  const int NT   = in_sizes[0] / 64;   // 200000
  const int NL   = in_sizes[8];        // 100000
  const long long E_tl = in_sizes[9];
  const long long E_lt = in_sizes[11];
  const long long E_ll = in_sizes[13];

  // Workspace layout (floats): agg_tl[NL*64] agg_ll[NL*64] agg_lt[NT*64]
  //                            cnt_tl[NL] cnt_ll[NL] cnt_lt[NT]
  float* agg_tl = (float*)d_ws;
  float* agg_ll = agg_tl + (size_t)NL * 64;
  float* agg_lt = agg_ll + (size_t)NL * 64;
  float* cnt_tl = agg_lt + (size_t)NT * 64;
  float* cnt_ll = cnt_tl + NL;
  float* cnt_lt = cnt_ll + NL;
  const size_t zero_bytes =
      ((size_t)NL * 128 + (size_t)NT * 64 + 2 * (size_t)NL + (size_t)NT) * sizeof(float);
  hipMemsetAsync(d_ws, 0, zero_bytes, stream);

  // --- edge aggregation (memory/atomic bound; L2-resident accumulators) ---
  const int TB = 256;
  auto blocks_for = [](long long work) -> unsigned int {
    long long b = (work + 255) / 256;
    if (b > (1LL << 22)) b = (1LL << 22);
    return (unsigned int)b;
  };
  gnn_edge_scatter<<<blocks_for(E_tl << 4), TB, 0, stream>>>(
      title_x, tl_src, tl_dst, nullptr, agg_tl, cnt_tl, E_tl);
  gnn_edge_scatter<<<blocks_for(E_ll << 4), TB, 0, stream>>>(
      label_embed, ll_src, ll_dst, label_node_id, agg_ll, cnt_ll, E_ll);
  gnn_edge_scatter<<<blocks_for(E_lt << 4), TB, 0, stream>>>(
      label_embed, lt_src, lt_dst, label_node_id, agg_lt, cnt_lt, E_lt);

  // --- WMMA epilogue GEMMs + bias + relu ---
  float* out_label = (float*)d_out;
  float* out_title = out_label + (size_t)NL * 64;

  const int WAVES_PER_BLOCK = 8;              // 256 threads, wave32
  const int label_tiles = (NL + 15) / 16;
  const int title_tiles = (NT + 15) / 16;
  gnn_label_out<<<(label_tiles + WAVES_PER_BLOCK - 1) / WAVES_PER_BLOCK,
                  WAVES_PER_BLOCK * 32, 0, stream>>>(
      label_embed, label_node_id, agg_tl, cnt_tl, agg_ll, cnt_ll,
      Wl_tl, bl_tl, Wr_tl, Wl_ll, bl_ll, Wr_ll, out_label, NL);
  gnn_title_out<<<(title_tiles + WAVES_PER_BLOCK - 1) / WAVES_PER_BLOCK,
                  WAVES_PER_BLOCK * 32, 0, stream>>>(
      title_x, agg_lt, cnt_lt, Wl_tl, bl_tl, Wr_tl, out_title, NT);
}